// LearnableEMA_14396730377045
// MI455X (gfx1250) — compile-verified
//
#include <hip/hip_runtime.h>
#include <stdint.h>

// Problem constants (from reference): x is (B, T, C) fp32.
#define B_  16
#define T_  4096
#define C_  512
#define K_  64            // chunks per (b,c) chain
#define L_  64            // timesteps per chunk  (K_ * L_ == T_)
#define TT_ 16            // timesteps per LDS tile
#define NT_ (L_ / TT_)    // tiles per chunk

#define CLAMP_LO 1.0e-4f
#define CLAMP_HI (1.0f - 1.0e-4f)

// ---------------------------------------------------------------------------
// CDNA5 async global->LDS copies (tracked by ASYNCcnt), per-lane b128.
// VDST = LDS byte address (low 32 bits of flat LDS pointer), VADDR = 64-bit
// global address. EXEC has all lanes active at every issue site.
//   _rt: default temporal hint  (pass 1: keep x resident in the 192MB L2)
//   _lu: last-use hint          (pass 3: x line is dead after this read)
// ---------------------------------------------------------------------------
__device__ __forceinline__ void async_copy_b128_rt(unsigned lds_addr, uint64_t gaddr) {
    asm volatile("global_load_async_to_lds_b128 %0, %1, off"
                 :: "v"(lds_addr), "v"(gaddr)
                 : "memory");
}

__device__ __forceinline__ void async_copy_b128_lu(unsigned lds_addr, uint64_t gaddr) {
    asm volatile("global_load_async_to_lds_b128 %0, %1, off th:TH_LOAD_LU"
                 :: "v"(lds_addr), "v"(gaddr)
                 : "memory");
}

__device__ __forceinline__ void wait_asynccnt_le4() {
#if __has_builtin(__builtin_amdgcn_s_wait_asynccnt)
    __builtin_amdgcn_s_wait_asynccnt(4);
#else
    asm volatile("s_wait_asynccnt 4" ::: "memory");
#endif
}

__device__ __forceinline__ void wait_asynccnt_0() {
#if __has_builtin(__builtin_amdgcn_s_wait_asynccnt)
    __builtin_amdgcn_s_wait_asynccnt(0);
#else
    asm volatile("s_wait_asynccnt 0" ::: "memory");
#endif
}

__device__ __forceinline__ float sigmoid_clamped(float z) {
    float s = 1.0f / (1.0f + expf(-z));
    return fminf(fmaxf(s, CLAMP_LO), CLAMP_HI);
}

// ---------------------------------------------------------------------------
// Pass 1 / Pass 3: per-chunk scan, streamed through LDS with double-buffered
// async copies. One block per (b, chunk); threadIdx.x == channel c.
//   WRITE_Y=false: start from s=0, emit chunk-end carry.
//   WRITE_Y=true : start from s=prefix[b,k,c], emit y for every timestep
//                  with non-temporal stores (y is never re-read; don't evict
//                  x from L2).
// Each tile (TT_ rows x C_ channels = 32 KB) is contiguous in global memory,
// so the copy is a flat b128 stream: 2048 b128 ops -> 4 per thread.
// ---------------------------------------------------------------------------
template <bool WRITE_Y>
__global__ __launch_bounds__(C_) void ema_chunk_kernel(
    const float* __restrict__ x,
    const float* __restrict__ logit_alpha,
    const float* __restrict__ prefix,   // [B][K][C] (only read when WRITE_Y)
    float* __restrict__ out)            // WRITE_Y ? y[B][T][C] : carry[B][K][C]
{
    __shared__ float smem[2][TT_ * C_];

    const int tid = threadIdx.x;             // channel
    const int b   = blockIdx.x / K_;
    const int k   = blockIdx.x % K_;
    const int t0  = k * L_;

    const float a = sigmoid_clamped(logit_alpha[tid]);
    const float w = 1.0f - a;

    // Chunk base: contiguous L_*C_ floats in global memory.
    const float* gbase = x + ((size_t)b * T_ + t0) * C_;

    const unsigned lds0 = (unsigned)(uintptr_t)&smem[0][0];
    const unsigned lds1 = (unsigned)(uintptr_t)&smem[1][0];

    // Prime the pipeline: tile 0 -> buffer 0.
    {
        const uint64_t g = (uint64_t)gbase;
        #pragma unroll
        for (int r = 0; r < 4; ++r) {
            const int i = tid + (r << 9);                  // b128 index
            const unsigned l = lds0 + (unsigned)(i << 4);
            const uint64_t ga = g + (uint64_t)(i << 4);
            if (WRITE_Y) async_copy_b128_lu(l, ga);
            else         async_copy_b128_rt(l, ga);
        }
    }

    float s = WRITE_Y ? prefix[((size_t)b * K_ + k) * C_ + tid] : 0.0f;

    for (int tile = 0; tile < NT_; ++tile) {
        if (tile + 1 < NT_) {
            // Prefetch next tile into the other buffer (safe: last readers of
            // that buffer finished before the barrier ending the previous
            // iteration), then wait until only those 4 new ops are in flight.
            const uint64_t g = (uint64_t)(gbase + (size_t)(tile + 1) * TT_ * C_);
            const unsigned lbase = ((tile + 1) & 1) ? lds1 : lds0;
            #pragma unroll
            for (int r = 0; r < 4; ++r) {
                const int i = tid + (r << 9);
                const unsigned l = lbase + (unsigned)(i << 4);
                const uint64_t ga = g + (uint64_t)(i << 4);
                if (WRITE_Y) async_copy_b128_lu(l, ga);
                else         async_copy_b128_rt(l, ga);
            }
            wait_asynccnt_le4();
        } else {
            wait_asynccnt_0();
        }
        __syncthreads();   // all waves' async writes for this tile now visible

        const float* buf = (tile & 1) ? &smem[1][0] : &smem[0][0];
        int gt = t0 + tile * TT_;
        #pragma unroll
        for (int row = 0; row < TT_; ++row, ++gt) {
            const float xv   = buf[row * C_ + tid];
            const float coef = (gt == 0) ? 1.0f : w;   // y0 = x0 exactly
            s = a * s + coef * xv;
            if (WRITE_Y) {
                __builtin_nontemporal_store(s, &out[((size_t)b * T_ + gt) * C_ + tid]);
            }
        }
        __syncthreads();   // done reading buf before it is overwritten
    }

    if (!WRITE_Y) {
        out[((size_t)b * K_ + k) * C_ + tid] = s;      // chunk-end carry
    }
}

// ---------------------------------------------------------------------------
// Pass 2: combine carries across chunks. One thread per (b,c) chain.
// prefix[b,k,c] = state entering chunk k (0 for k==0).
// ---------------------------------------------------------------------------
__global__ __launch_bounds__(256) void ema_prefix_kernel(
    const float* __restrict__ logit_alpha,
    const float* __restrict__ carry,    // [B][K][C]
    float* __restrict__ prefix)         // [B][K][C]
{
    const int idx = blockIdx.x * blockDim.x + threadIdx.x;
    if (idx >= B_ * C_) return;
    const int b = idx / C_;
    const int c = idx % C_;

    const float a = sigmoid_clamped(logit_alpha[c]);
    float aL = a;                       // a^64 via 6 squarings
    #pragma unroll
    for (int i = 0; i < 6; ++i) aL *= aL;

    float state = 0.0f;
    for (int k = 0; k < K_; ++k) {
        const size_t o = ((size_t)b * K_ + k) * C_ + c;
        prefix[o] = state;
        state = aL * state + carry[o];
    }
}

// ---------------------------------------------------------------------------
// Fallback (workspace too small): one thread per chain, plain serial scan.
// ---------------------------------------------------------------------------
__global__ __launch_bounds__(256) void ema_simple_kernel(
    const float* __restrict__ x,
    const float* __restrict__ logit_alpha,
    float* __restrict__ y)
{
    const int idx = blockIdx.x * blockDim.x + threadIdx.x;
    if (idx >= B_ * C_) return;
    const int b = idx / C_;
    const int c = idx % C_;

    const float a = sigmoid_clamped(logit_alpha[c]);
    const float w = 1.0f - a;

    const size_t base = (size_t)b * T_ * C_ + c;
    float s = x[base];
    y[base] = s;
    for (int t = 1; t < T_; ++t) {
        s = a * s + w * x[base + (size_t)t * C_];
        y[base + (size_t)t * C_] = s;
    }
}

extern "C" void kernel_launch(void* const* d_in, const int* in_sizes, int n_in,
                              void* d_out, int out_size, void* d_ws, size_t ws_size,
                              hipStream_t stream) {
    const float* x  = (const float*)d_in[0];   // (B, T, C) fp32
    const float* la = (const float*)d_in[1];   // (C,) fp32
    float* y = (float*)d_out;                  // (B, T, C) fp32

    const size_t chain_elems = (size_t)B_ * K_ * C_;           // 524288
    const size_t need = 2 * chain_elems * sizeof(float);       // 4 MB

    if (d_ws == nullptr || ws_size < need) {
        const int n = B_ * C_;
        ema_simple_kernel<<<(n + 255) / 256, 256, 0, stream>>>(x, la, y);
        return;
    }

    float* carry  = (float*)d_ws;
    float* prefix = carry + chain_elems;

    // Pass 1: chunk-local scans -> carries (streams x once, default RT hint so
    // x stays resident in the 192MB L2 for pass 3).
    ema_chunk_kernel<false><<<B_ * K_, C_, 0, stream>>>(x, la, nullptr, carry);

    // Pass 2: sequential combine of 64 carries per chain (2 MB in / 2 MB out).
    {
        const int n = B_ * C_;
        ema_prefix_kernel<<<(n + 255) / 256, 256, 0, stream>>>(la, carry, prefix);
    }

    // Pass 3: re-scan each chunk seeded with its prefix (last-use loads, since
    // x is dead afterwards), write y with non-temporal stores.
    ema_chunk_kernel<true><<<B_ * K_, C_, 0, stream>>>(x, la, prefix, y);
}